// GCN_9491877724720
// MI455X (gfx1250) — compile-verified
//
#include <hip/hip_runtime.h>
#include <hip/hip_bf16.h>

typedef __attribute__((ext_vector_type(2))) float v2f;
typedef __attribute__((ext_vector_type(8))) float v8f;

#define FEAT 64

// ---------------- utility kernels ----------------

__global__ void fill_f32(float* __restrict__ p, float v, int n) {
  int i = blockIdx.x * blockDim.x + threadIdx.x;
  if (i < n) p[i] = v;
}

__global__ void deg_accum(const int* __restrict__ dst, float* __restrict__ deg, int nE) {
  int e = blockIdx.x * blockDim.x + threadIdx.x;
  if (e < nE) atomicAdd(&deg[dst[e]], 1.0f);
}

__global__ void deg_finalize(float* __restrict__ deg, int n) {
  int i = blockIdx.x * blockDim.x + threadIdx.x;
  if (i < n) deg[i] = rsqrtf(deg[i]);  // deg already includes +1 self loop
}

// ---------------- WMMA GEMM: Out[N,64] = X[N,64] @ W[64,64] ----------------
// One wave32 computes a 16x64 strip of Out: 4 independent 16x16 accumulator
// chains over K=64 (16 steps of V_WMMA_F32_16X16X4_F32 each). fp32 in/out,
// so numerics match the reference. A fragments are reused across the 4 column
// tiles; the 4 WMMAs per k-step are independent, hiding WMMA->WMMA latency.
//
// A (16x4 f32) lane layout: lanes 0-15 -> M=lane, regs {K=0,K=1};
//                           lanes 16-31 -> M=lane-16, regs {K=2,K=3}.
// B (4x16 f32): lane holds col lane%16, regs {K=k0,K=k0+1}, k0=2*(lane/16).
// C/D (16x16 f32, 8 regs): reg r -> M = r + 8*(lane/16), N = lane%16.
__global__ void gemm_x64_wmma(const float* __restrict__ X,
                              const float* __restrict__ W,
                              float* __restrict__ Out, int nrows) {
  const int wavesPerBlock = blockDim.x >> 5;
  const int wave = blockIdx.x * wavesPerBlock + ((int)threadIdx.x >> 5);
  const int lane = threadIdx.x & 31;
  const int mTiles = (nrows + 15) >> 4;
  if (wave >= mTiles) return;        // uniform per-wave exit: EXEC stays all-1s
  const int m0 = wave << 4;

  const int half = lane >> 4;        // 0 or 1
  const int l16  = lane & 15;
  int row = m0 + l16;
  if (row >= nrows) row = nrows - 1; // clamp (no divergent loads); stores guarded
  const float* __restrict__ xr = X + (size_t)row * FEAT;

  v8f acc0 = {}, acc1 = {}, acc2 = {}, acc3 = {};
#pragma unroll
  for (int kt = 0; kt < 16; ++kt) {
    const int k0 = (kt << 2) + (half << 1);
    const float2 av = *(const float2*)(xr + k0);   // one b64 load
    v2f a; a.x = av.x; a.y = av.y;
    const float* __restrict__ wr0 = W + (size_t)k0 * FEAT + l16;
    const float* __restrict__ wr1 = wr0 + FEAT;
    v2f b0, b1, b2, b3;
    b0.x = wr0[0];  b0.y = wr1[0];
    b1.x = wr0[16]; b1.y = wr1[16];
    b2.x = wr0[32]; b2.y = wr1[32];
    b3.x = wr0[48]; b3.y = wr1[48];
    acc0 = __builtin_amdgcn_wmma_f32_16x16x4_f32(false, a, false, b0, (short)0, acc0, false, false);
    acc1 = __builtin_amdgcn_wmma_f32_16x16x4_f32(false, a, false, b1, (short)0, acc1, false, false);
    acc2 = __builtin_amdgcn_wmma_f32_16x16x4_f32(false, a, false, b2, (short)0, acc2, false, false);
    acc3 = __builtin_amdgcn_wmma_f32_16x16x4_f32(false, a, false, b3, (short)0, acc3, false, false);
  }

  const int mBase = m0 + (half << 3);
#pragma unroll
  for (int r = 0; r < 8; ++r) {
    const int m = mBase + r;
    if (m < nrows) {
      float* __restrict__ orow = Out + (size_t)m * FEAT + l16;
      orow[0]  = acc0[r];
      orow[16] = acc1[r];
      orow[32] = acc2[r];
      orow[48] = acc3[r];
    }
  }
}

// ---------------- edge scatter: agg[dst] += dinv[src]*dinv[dst]*H[src] ----
// 16 lanes per edge, one b128 row-chunk load each, 4 f32 atomics. Both H and
// agg (25.6MB each) are resident in the 192MB L2, so the random-access
// atomics resolve on-chip; this kernel is the runtime roofline limiter.
__global__ void scatter_edges(const float* __restrict__ H,
                              const int* __restrict__ src,
                              const int* __restrict__ dst,
                              const float* __restrict__ dinv,
                              float* __restrict__ agg, int nE) {
  unsigned idx = blockIdx.x * blockDim.x + threadIdx.x;
  unsigned e = idx >> 4;
  unsigned j = idx & 15;
  if (e >= (unsigned)nE) return;
  const int s = src[e];
  const int d = dst[e];
  const float coef = dinv[s] * dinv[d];
  const float4 hv = *(const float4*)(H + (size_t)s * FEAT + j * 4u);
  float* ap = agg + (size_t)d * FEAT + j * 4u;
  atomicAdd(ap + 0, coef * hv.x);
  atomicAdd(ap + 1, coef * hv.y);
  atomicAdd(ap + 2, coef * hv.z);
  atomicAdd(ap + 3, coef * hv.w);
}

// ---------------- self term + bias (+ relu): agg += dinv^2 * T + b --------
__global__ void self_bias_relu(float* __restrict__ agg, const float* __restrict__ T,
                               const float* __restrict__ dinv,
                               const float* __restrict__ bias,
                               int nNodes, int doRelu) {
  int idx = blockIdx.x * blockDim.x + threadIdx.x;
  int i = idx >> 6, f = idx & 63;
  if (i >= nNodes) return;
  const float di = dinv[i];
  float v = agg[(size_t)i * FEAT + f] + di * di * T[(size_t)i * FEAT + f] + bias[f];
  agg[(size_t)i * FEAT + f] = doRelu ? fmaxf(v, 0.0f) : v;
}

// ---------------- mean pool accumulate ----------------
__global__ void pool_accum(const float* __restrict__ H, const int* __restrict__ batch,
                           float* __restrict__ sums, float* __restrict__ cnts, int nNodes) {
  int idx = blockIdx.x * blockDim.x + threadIdx.x;
  int i = idx >> 6, f = idx & 63;
  if (i >= nNodes) return;
  const int g = batch[i];
  atomicAdd(&sums[g * FEAT + f], H[(size_t)i * FEAT + f]);
  if (f == 0) atomicAdd(&cnts[g], 1.0f);
}

// ---------------- final linear: out[g,c] = mean(g) . Wl[:,c] + bl[c] ------
__global__ void final_linear(const float* __restrict__ sums, const float* __restrict__ cnts,
                             const float* __restrict__ Wl, const float* __restrict__ bl,
                             float* __restrict__ out, int nGraphs, int nClasses) {
  int t = blockIdx.x * blockDim.x + threadIdx.x;
  if (t >= nGraphs * nClasses) return;
  const int g = t / nClasses, c = t % nClasses;
  const float inv = 1.0f / fmaxf(cnts[g], 1.0f);
  float acc = bl[c];
  for (int k = 0; k < FEAT; ++k)
    acc += (sums[g * FEAT + k] * inv) * Wl[k * nClasses + c];
  out[t] = acc;
}

// ---------------- host orchestration ----------------
extern "C" void kernel_launch(void* const* d_in, const int* in_sizes, int n_in,
                              void* d_out, int out_size, void* d_ws, size_t ws_size,
                              hipStream_t stream) {
  const float* x     = (const float*)d_in[0];
  const int*   ei    = (const int*)  d_in[1];
  const int*   batch = (const int*)  d_in[2];
  const float* W1 = (const float*)d_in[3];
  const float* b1 = (const float*)d_in[4];
  const float* W2 = (const float*)d_in[5];
  const float* b2 = (const float*)d_in[6];
  const float* W3 = (const float*)d_in[7];
  const float* b3 = (const float*)d_in[8];
  const float* Wl = (const float*)d_in[9];
  const float* bl = (const float*)d_in[10];
  float* out = (float*)d_out;

  const int nNodes   = in_sizes[0] / FEAT;
  const int nEdges   = in_sizes[1] / 2;
  const int nClasses = in_sizes[10];           // = 2
  const int nGraphs  = out_size / nClasses;    // = 64

  const int* src  = ei;              // edge_index row 0
  const int* dstp = ei + nEdges;     // edge_index row 1

  // workspace layout (floats): dinv | T (N*64) | Hb (N*64) | sums | cnts
  float* dinv = (float*)d_ws;
  float* T    = dinv + nNodes;
  float* Hb   = T + (size_t)nNodes * FEAT;
  float* sums = Hb + (size_t)nNodes * FEAT;
  float* cnts = sums + (size_t)nGraphs * FEAT;

  const int TPB = 256;
  auto cdiv = [](long long a, long long b) { return (int)((a + b - 1) / b); };

  // degrees of A+I, then dinv = rsqrt(deg)
  fill_f32<<<cdiv(nNodes, TPB), TPB, 0, stream>>>(dinv, 1.0f, nNodes);
  deg_accum<<<cdiv(nEdges, TPB), TPB, 0, stream>>>(dstp, dinv, nEdges);
  deg_finalize<<<cdiv(nNodes, TPB), TPB, 0, stream>>>(dinv, nNodes);

  const long long nodeFeat = (long long)nNodes * FEAT;
  const long long edgeWork = (long long)nEdges * 16;
  const int mTiles = (nNodes + 15) / 16;
  const int gemmBlocks = cdiv(mTiles, TPB / 32);   // one wave per 16-row strip

  const float* curIn = x;
  const float* Ws[3] = {W1, W2, W3};
  const float* bs[3] = {b1, b2, b3};
  for (int layer = 0; layer < 3; ++layer) {
    gemm_x64_wmma<<<gemmBlocks, TPB, 0, stream>>>(curIn, Ws[layer], T, nNodes);
    fill_f32<<<cdiv(nodeFeat, TPB), TPB, 0, stream>>>(Hb, 0.0f, (int)nodeFeat);
    scatter_edges<<<cdiv(edgeWork, TPB), TPB, 0, stream>>>(T, src, dstp, dinv, Hb, nEdges);
    self_bias_relu<<<cdiv(nodeFeat, TPB), TPB, 0, stream>>>(Hb, T, dinv, bs[layer],
                                                            nNodes, layer < 2 ? 1 : 0);
    curIn = Hb;
  }

  fill_f32<<<cdiv(nGraphs * FEAT + nGraphs, TPB), TPB, 0, stream>>>(
      sums, 0.0f, nGraphs * FEAT + nGraphs);
  pool_accum<<<cdiv(nodeFeat, TPB), TPB, 0, stream>>>(Hb, batch, sums, cnts, nNodes);
  final_linear<<<cdiv(nGraphs * nClasses, TPB), TPB, 0, stream>>>(
      sums, cnts, Wl, bl, out, nGraphs, nClasses);
}